// Involuted_ToroidalSOM_4020089389312
// MI455X (gfx1250) — compile-verified
//
#include <hip/hip_runtime.h>
#include <hip/hip_bf16.h>

// ---------------------------------------------------------------------------
// Involuted Toroidal SOM step for MI455X (gfx1250, wave32).
//
//   x        : [512, 64]  f32
//   weights  : [128,128,64] f32  (16384 units x 64 dims)
//   bmu_usage: [128,128]  f32
//   out      : weights + (lr/B) * (neigh^T @ x - rowsum(neigh) * weights)
//
// Pipeline (one stream, serialized):
//   0) max(bmu_usage)                               -> ws_max
//   1) BMU partial argmin via f32 WMMA GEMM (32x8)  -> ws_pb / ws_pi
//   2) final argmin across unit slices              -> ws_bmu
//   3) fused neigh + scatter-GEMM + update          -> out
//      (X staged in LDS via global_load_async_to_lds_b128 / ASYNCcnt)
// ---------------------------------------------------------------------------

typedef float v8f __attribute__((ext_vector_type(8)));
typedef float v2f __attribute__((ext_vector_type(2)));

#define B_SZ   512
#define GRID_R 128
#define GRID_C 128
#define N_UNIT (GRID_R * GRID_C)   // 16384
#define D_SZ   64
#define SIGMA  3.0f
#define LR_OVER_B (0.1f / 512.0f)
#define MASK_R2 20.25f             // (sigma*1.5)^2
#define N_SLICE 8                  // unit-axis slices for the BMU kernel

// ---------------------------------------------------------------- kernel 0 --
__global__ void maxu_kernel(const float* __restrict__ usage,
                            float* __restrict__ outmax) {
    __shared__ float red[8];
    float m = -3.4e38f;
    for (int i = threadIdx.x; i < N_UNIT; i += 256)
        m = fmaxf(m, usage[i]);
    #pragma unroll
    for (int off = 16; off >= 1; off >>= 1)
        m = fmaxf(m, __shfl_xor(m, off, 32));
    if ((threadIdx.x & 31) == 0) red[threadIdx.x >> 5] = m;
    __syncthreads();
    if (threadIdx.x == 0) {
        float r = red[0];
        #pragma unroll
        for (int i = 1; i < 8; ++i) r = fmaxf(r, red[i]);
        outmax[0] = r;
    }
}

// ---------------------------------------------------------------- kernel 1 --
// grid (32 batch tiles x 8 unit slices), 4 waves/block. Each wave sweeps 32
// unit tiles of its slice. Distance tile D[16b x 16u] = ||w||^2 - 2*x.w
// (the ||x||^2 term cannot change the argmin). Per tile: 16 b64 loads issued
// up-front into distinct regs, then two independent WMMA chains (even/odd K).
__global__ void bmu_part_kernel(const float* __restrict__ x,
                                const float* __restrict__ w,
                                float* __restrict__ pbest,
                                int* __restrict__ pidx) {
    const int tid  = threadIdx.x;           // 128 threads
    const int wave = tid >> 5;
    const int lane = tid & 31;
    const int ln   = lane & 15;             // N-column / M-row lane index
    const int half = lane >> 4;             // 0: K+{0,1}, 1: K+{2,3}
    const int b_base = blockIdx.x * 16;
    const int slice  = blockIdx.y;

    // Whole 16x64 X tile kept in registers as 16 WMMA A-operands
    // (f32 16x4 layout: lanes 0-15 M=row, vgpr0 K=4k+2h, vgpr1 K=4k+2h+1).
    v2f a[16];
    {
        const float* ap = x + (b_base + ln) * D_SZ + half * 2;
        #pragma unroll
        for (int kk = 0; kk < 16; ++kk)
            a[kk] = *(const v2f*)(ap + kk * 4);
    }

    float best[8];
    int   bidx[8];
    #pragma unroll
    for (int i = 0; i < 8; ++i) { best[i] = 3.4e38f; bidx[i] = 0x7fffffff; }

    const int tile0 = slice * (N_UNIT / 16 / N_SLICE);   // 128 tiles per slice
    for (int t = 0; t < 32; ++t) {
        const int u0 = (tile0 + wave + 4 * t) * 16;
        const float* wrow = w + (u0 + ln) * D_SZ + half * 2;
        __builtin_prefetch(wrow + 4 * 16 * D_SZ, 0, 3);  // this wave's next tile

        // batch the 16 B-operand loads so they overlap (distinct dest regs)
        v2f bv[16];
        #pragma unroll
        for (int kk = 0; kk < 16; ++kk)
            bv[kk] = *(const v2f*)(wrow + kk * 4);

        v8f acc_e = {0.f,0.f,0.f,0.f,0.f,0.f,0.f,0.f};
        v8f acc_o = acc_e;
        float w2p = 0.f;
        #pragma unroll
        for (int kk = 0; kk < 16; kk += 2) {
            w2p += bv[kk][0] * bv[kk][0] + bv[kk][1] * bv[kk][1];
            w2p += bv[kk+1][0] * bv[kk+1][0] + bv[kk+1][1] * bv[kk+1][1];
            acc_e = __builtin_amdgcn_wmma_f32_16x16x4_f32(
                        false, a[kk],   false, bv[kk],   (short)0, acc_e, false, false);
            acc_o = __builtin_amdgcn_wmma_f32_16x16x4_f32(
                        false, a[kk+1], false, bv[kk+1], (short)0, acc_o, false, false);
        }
        // each lane saw half the K range of its unit; partner lane (^16) has the rest
        const float w2f = w2p + __shfl_xor(w2p, 16, 32);
        const int uidx = u0 + ln;
        #pragma unroll
        for (int i = 0; i < 8; ++i) {
            const float dist = w2f - 2.0f * (acc_e[i] + acc_o[i]); // row M=i+8*half
            if (dist < best[i] || (dist == best[i] && uidx < bidx[i])) {
                best[i] = dist; bidx[i] = uidx;
            }
        }
    }

    // argmin across the 16 column-lanes of each half-wave
    #pragma unroll
    for (int i = 0; i < 8; ++i) {
        #pragma unroll
        for (int off = 8; off >= 1; off >>= 1) {
            const float ov = __shfl_xor(best[i], off, 32);
            const int   oi = __shfl_xor(bidx[i], off, 32);
            if (ov < best[i] || (ov == best[i] && oi < bidx[i])) {
                best[i] = ov; bidx[i] = oi;
            }
        }
    }

    __shared__ float s_best[4][16];
    __shared__ int   s_idx[4][16];
    if (ln == 0) {          // lanes 0 and 16 carry rows i and i+8
        #pragma unroll
        for (int i = 0; i < 8; ++i) {
            s_best[wave][i + 8 * half] = best[i];
            s_idx [wave][i + 8 * half] = bidx[i];
        }
    }
    __syncthreads();
    if (tid < 16) {
        float bv2 = s_best[0][tid]; int bi = s_idx[0][tid];
        #pragma unroll
        for (int wv = 1; wv < 4; ++wv) {
            const float ov = s_best[wv][tid]; const int oi = s_idx[wv][tid];
            if (ov < bv2 || (ov == bv2 && oi < bi)) { bv2 = ov; bi = oi; }
        }
        const int slot = (blockIdx.x * N_SLICE + slice) * 16 + tid;
        pbest[slot] = bv2;
        pidx [slot] = bi;
    }
}

// ---------------------------------------------------------------- kernel 2 --
__global__ void bmu_final_kernel(const float* __restrict__ pbest,
                                 const int* __restrict__ pidx,
                                 int* __restrict__ bmu) {
    const int b = blockIdx.x * 256 + threadIdx.x;   // 0..511
    const int bt = b >> 4, r = b & 15;
    float bv = 3.4e38f; int bi = 0x7fffffff;
    #pragma unroll
    for (int sl = 0; sl < N_SLICE; ++sl) {
        const int slot = (bt * N_SLICE + sl) * 16 + r;
        const float ov = pbest[slot];
        const int   oi = pidx[slot];
        if (ov < bv || (ov == bv && oi < bi)) { bv = ov; bi = oi; }
    }
    bmu[b] = bi;
}

// ------------------------------------------------------------- neigh math --
__device__ __forceinline__ float neigh_val(int rm, int cm, int br, int bc,
                                           float ab) {
    int rd = rm - br; rd = rd < 0 ? -rd : rd; rd = rd < GRID_R - rd ? rd : GRID_R - rd;
    int cd = cm - bc; cd = cd < 0 ? -cd : cd; cd = cd < GRID_C - cd ? cd : GRID_C - cd;
    const float rdf = (float)rd, cdf = (float)cd;
    const float hex = cdf + 0.5f * (float)(rd & 1);
    const float rscaled = rdf * 0.866f;
    const float dist_sq = hex * hex + rscaled * rscaled;
    const float r2  = cdf * cdf + rdf * rdf;
    const float r_s = __fsqrt_rn(r2) * (1.0f / SIGMA);
    const float center = ab * __expf(-r_s * r_s);
    const float t = r_s - 2.0f * ab;
    const float ring = __expf(-t * t * 0.25f);      // / BUMP_W^2 = /4
    float v = ring - center;
    v = v > 0.f ? v : 0.f;
    return (dist_sq < MASK_R2) ? v : 0.f;
}

// ---------------------------------------------------------------- kernel 3 --
// out[u,d] = w[u,d] + (lr/B)*(sum_b neigh[b,u]*x[b,d] - (sum_b neigh[b,u])*w[u,d])
// 256 blocks x 4 waves; each wave owns one 16-unit tile, K=512 in 128 WMMA
// k-steps, 4 D-tiles per step, A = neigh computed in-register in WMMA layout.
// X (128 KB) is staged once per block into LDS with the gfx1250 async-to-LDS
// path (each wave in the block would otherwise stream the full X from L2).
__global__ void update_kernel(const float* __restrict__ x,
                              const float* __restrict__ w,
                              const float* __restrict__ usage,
                              const int* __restrict__ bmu,
                              const float* __restrict__ maxu_p,
                              float* __restrict__ out) {
    __shared__ float s_x[B_SZ * D_SZ];   // 128 KB: full X, LDS-resident
    __shared__ int   s_br[B_SZ];
    __shared__ int   s_bc[B_SZ];
    __shared__ float s_ab[B_SZ];

    const int tid = threadIdx.x;            // 128 threads

    // ---- async copy X -> LDS (16 B per lane per issue, ASYNCcnt-tracked) ----
    {
        const unsigned chunks = (B_SZ * D_SZ * 4) / 16;      // 8192 x b128
        const char* gbase = (const char*)x;
        for (unsigned c = tid; c < chunks; c += 128) {
            const unsigned byte_off = c * 16u;
            const unsigned lds_addr =
                (unsigned)(size_t)((char*)s_x + byte_off);   // low 32 bits = LDS offset
            const unsigned long long gaddr =
                (unsigned long long)(size_t)(gbase + byte_off);
            asm volatile("global_load_async_to_lds_b128 %0, %1, off"
                         :: "v"(lds_addr), "v"(gaddr) : "memory");
        }
        asm volatile("s_wait_asynccnt 0x0" ::: "memory");
    }

    // ---- per-batch BMU metadata -> LDS ----
    const float inv = 1.0f / (maxu_p[0] + 1e-8f);
    for (int b = tid; b < B_SZ; b += 128) {
        const int idx = bmu[b];
        s_br[b] = idx >> 7;
        s_bc[b] = idx & (GRID_C - 1);
        s_ab[b] = usage[idx] * inv;
    }
    __syncthreads();

    const int lane = tid & 31;
    const int wave = tid >> 5;
    const int ln   = lane & 15;
    const int half = lane >> 4;
    const int u0   = (blockIdx.x * 4 + wave) * 16;
    const int rm   = u0 >> 7;               // tile never crosses a grid row
    const int cm   = (u0 & (GRID_C - 1)) + ln;

    // warm the W tile we will read at the end (fused into the store phase)
    __builtin_prefetch(w + u0 * D_SZ + lane * 32, 0, 3);

    v8f acc0 = {0.f,0.f,0.f,0.f,0.f,0.f,0.f,0.f};
    v8f acc1 = acc0, acc2 = acc0, acc3 = acc0;
    float ssum = 0.f;                       // per-lane partial of sum_b neigh

    for (int kk = 0; kk < B_SZ / 4; ++kk) {
        const int k0 = kk * 4 + half * 2;   // this lane's two K slots (A layout)

        // B = X rows k0..k0+1 (from LDS), 4 D-tiles of 16; issue the DS loads
        // first so they hide under the transcendental-heavy neigh chain.
        const float* xr = s_x + k0 * D_SZ + ln;
        v2f b0; b0[0] = xr[0];  b0[1] = xr[D_SZ + 0];
        v2f b1; b1[0] = xr[16]; b1[1] = xr[D_SZ + 16];
        v2f b2; b2[0] = xr[32]; b2[1] = xr[D_SZ + 32];
        v2f b3; b3[0] = xr[48]; b3[1] = xr[D_SZ + 48];

        const float na = neigh_val(rm, cm, s_br[k0],     s_bc[k0],     s_ab[k0]);
        const float nb = neigh_val(rm, cm, s_br[k0 + 1], s_bc[k0 + 1], s_ab[k0 + 1]);
        ssum += na + nb;
        v2f av; av[0] = na; av[1] = nb;

        acc0 = __builtin_amdgcn_wmma_f32_16x16x4_f32(false, av, false, b0, (short)0, acc0, false, false);
        acc1 = __builtin_amdgcn_wmma_f32_16x16x4_f32(false, av, false, b1, (short)0, acc1, false, false);
        acc2 = __builtin_amdgcn_wmma_f32_16x16x4_f32(false, av, false, b2, (short)0, acc2, false, false);
        acc3 = __builtin_amdgcn_wmma_f32_16x16x4_f32(false, av, false, b3, (short)0, acc3, false, false);
    }

    // total neighborhood mass for unit m = ln (partner half has the other K's)
    const float s = ssum + __shfl_xor(ssum, 16, 32);
    // redistribute: C-row M = i + 8*half needs s[M], which lives in lane M%16
    float sM[8];
    #pragma unroll
    for (int i = 0; i < 8; ++i)
        sM[i] = __shfl(s, i + half * 8, 32);

    #pragma unroll
    for (int i = 0; i < 8; ++i) {
        const int u = u0 + i + 8 * half;
        const int base = u * D_SZ + ln;
        const float sv = sM[i];
        const float w0 = w[base +  0], w1 = w[base + 16];
        const float w2 = w[base + 32], w3 = w[base + 48];
        out[base +  0] = w0 + LR_OVER_B * (acc0[i] - sv * w0);
        out[base + 16] = w1 + LR_OVER_B * (acc1[i] - sv * w1);
        out[base + 32] = w2 + LR_OVER_B * (acc2[i] - sv * w2);
        out[base + 48] = w3 + LR_OVER_B * (acc3[i] - sv * w3);
    }
}

// --------------------------------------------------------------- launcher --
extern "C" void kernel_launch(void* const* d_in, const int* in_sizes, int n_in,
                              void* d_out, int out_size, void* d_ws, size_t ws_size,
                              hipStream_t stream) {
    const float* x     = (const float*)d_in[0];   // [512,64]
    const float* wts   = (const float*)d_in[1];   // [128,128,64]
    const float* usage = (const float*)d_in[2];   // [128,128]
    float* out = (float*)d_out;

    char* ws = (char*)d_ws;
    float* ws_max = (float*)ws;                   // 1 float        @ 0
    int*   ws_bmu = (int*)(ws + 64);              // 512 ints       @ 64
    float* ws_pb  = (float*)(ws + 4096);          // 32*8*16 floats @ 4 KB
    int*   ws_pi  = (int*)(ws + 20480);           // 32*8*16 ints   @ 20 KB

    maxu_kernel     <<<1, 256, 0, stream>>>(usage, ws_max);
    bmu_part_kernel <<<dim3(32, N_SLICE), 128, 0, stream>>>(x, wts, ws_pb, ws_pi);
    bmu_final_kernel<<<2, 256, 0, stream>>>(ws_pb, ws_pi, ws_bmu);
    update_kernel   <<<256, 128, 0, stream>>>(x, wts, usage, ws_bmu, ws_max, out);
}